// MSELoss_52905407152940
// MI455X (gfx1250) — compile-verified
//
#include <hip/hip_runtime.h>

typedef __attribute__((ext_vector_type(2))) float v2f;
typedef __attribute__((ext_vector_type(8))) float v8f;

// Problem constants from the reference: B=8, C=4, N=16, H=W=256
#define HW4     16384        // dwords (or float4 groups) per H*W plane
#define NSLOT   16
#define CH      4
#define NBLOCKS 512
#define NTHREADS 256
// NBLOCKS*NTHREADS == 131072 == number of 4-pixel groups (8*65536/4) exactly.

__device__ __forceinline__ float elem4(const float4& v, int j) {
    return ((const float*)&v)[j];
}

// ds_swizzle_b32 xor-swap within 32-lane group: offset[15]=0,
// offset = {xor_mask[4:0] << 10, or_mask=0 << 5, and_mask=0x1f}
template <int PATT>
__device__ __forceinline__ float swz_xor(float x) {
    return __int_as_float(__builtin_amdgcn_ds_swizzle(__float_as_int(x), PATT));
}

__global__ __launch_bounds__(NTHREADS) void mse_main_kernel(
    const float* __restrict__ pd,
    const float* __restrict__ gt,
    const unsigned char* __restrict__ pd_mask,
    const unsigned char* __restrict__ gt_mask,
    float* __restrict__ partials)
{
    const unsigned int gid = blockIdx.x * NTHREADS + threadIdx.x; // 0..131071
    const unsigned int b   = gid >> 14;      // batch index (16384 dword-groups/plane-set)
    const unsigned int dw  = gid & 16383;    // dword index within an H*W plane

    // ---- mask counts: 4 pixels per lane, packed in byte lanes ----
    const unsigned int* pm32 = (const unsigned int*)pd_mask;
    const unsigned int* gm32 = (const unsigned int*)gt_mask;
    const unsigned int mbase = b * (NSLOT * HW4) + dw;

    unsigned int sumP = 0u, sumG = 0u, sumPG = 0u;
#pragma unroll
    for (int n = 0; n < NSLOT; ++n) {
        unsigned int pmv = pm32[mbase + n * HW4];
        unsigned int gmv = gm32[mbase + n * HW4];
        sumP  += pmv;
        sumG  += gmv;
        sumPG += (pmv & gmv);
    }

    // ---- per-channel float data: float4 = the same 4 pixels ----
    const float4* pd4 = (const float4*)pd;
    const float4* gt4 = (const float4*)gt;
    const unsigned int fbase = b * (CH * HW4) + dw;

    float4 p0 = pd4[fbase + 0 * HW4];
    float4 p1 = pd4[fbase + 1 * HW4];
    float4 p2 = pd4[fbase + 2 * HW4];
    float4 p3 = pd4[fbase + 3 * HW4];
    float4 g0 = gt4[fbase + 0 * HW4];
    float4 g1 = gt4[fbase + 1 * HW4];
    float4 g2 = gt4[fbase + 2 * HW4];
    float4 g3 = gt4[fbase + 3 * HW4];

    float acc = 0.0f;
#pragma unroll
    for (int j = 0; j < 4; ++j) {
        float a0 = elem4(p0, j), a1 = elem4(p1, j), a2 = elem4(p2, j), a3 = elem4(p3, j);
        float b0 = elem4(g0, j), b1 = elem4(g1, j), b2 = elem4(g2, j), b3 = elem4(g3, j);

        float spd2 = a0 * a0 + a1 * a1 + a2 * a2 + a3 * a3;   // sum_c pd^2
        float sgt2 = b0 * b0 + b1 * b1 + b2 * b2 + b3 * b3;   // sum_c gt^2
        float spg  = a0 * b0 + a1 * b1 + a2 * b2 + a3 * b3;   // sum_c pd*gt

        float cp  = (float)((sumP  >> (8 * j)) & 0xFFu);
        float cg  = (float)((sumG  >> (8 * j)) & 0xFFu);
        float cpg = (float)((sumPG >> (8 * j)) & 0xFFu);

        acc += spd2 * cp + sgt2 * cg - 2.0f * spg * cpg;
    }

    // ---- wave32 reduction via V_WMMA_F32_16X16X4_F32 with A = ones ----
    // D[m,n] = sum_k B[k,n] for every m. Each lane deposits `acc` into both
    // B VGPRs, so the 16 column sums (held in comp0 of lanes 0-15, duplicated
    // in lanes 16-31) add up to 2 * (wave total), independent of the exact
    // (k,n) <-> (vgpr,lane) bijection.
    v2f aOnes; aOnes[0] = 1.0f; aOnes[1] = 1.0f;
    v2f bVal;  bVal[0]  = acc;  bVal[1]  = acc;
    v8f cZero = {};
    v8f dmat = __builtin_amdgcn_wmma_f32_16x16x4_f32(
        false, aOnes, false, bVal, (short)0, cZero, false, false);

    // 16-lane xor-reduce (both halves reduce symmetrically): 4 swizzle steps.
    float wsum = dmat[0];
    wsum += swz_xor<0x201f>(wsum); // xor 8
    wsum += swz_xor<0x101f>(wsum); // xor 4
    wsum += swz_xor<0x081f>(wsum); // xor 2
    wsum += swz_xor<0x041f>(wsum); // xor 1
    wsum *= 0.5f; // column sums totaled 2x the wave sum

    // ---- cross-wave (8 waves/block) reduction in LDS ----
    __shared__ float red[NTHREADS / 32];
    const int wave = threadIdx.x >> 5;
    const int lane = threadIdx.x & 31;
    if (lane == 0) red[wave] = wsum;
    __syncthreads();
    if (threadIdx.x == 0) {
        float s = 0.0f;
#pragma unroll
        for (int w = 0; w < NTHREADS / 32; ++w) s += red[w];
        partials[blockIdx.x] = s;
    }
}

__global__ __launch_bounds__(512) void mse_finish_kernel(
    const float* __restrict__ partials,
    float* __restrict__ out)
{
    __shared__ float red[16];
    float v = partials[threadIdx.x]; // exactly NBLOCKS = 512 partials
    v += swz_xor<0x401f>(v); // xor 16
    v += swz_xor<0x201f>(v); // xor 8
    v += swz_xor<0x101f>(v); // xor 4
    v += swz_xor<0x081f>(v); // xor 2
    v += swz_xor<0x041f>(v); // xor 1
    if ((threadIdx.x & 31) == 0) red[threadIdx.x >> 5] = v;
    __syncthreads();
    if (threadIdx.x == 0) {
        float s = 0.0f;
#pragma unroll
        for (int i = 0; i < 16; ++i) s += red[i];
        out[0] = s * (1.0f / 33554432.0f); // mean over B*N*C*H*W = 8*16*4*256*256
    }
}

extern "C" void kernel_launch(void* const* d_in, const int* in_sizes, int n_in,
                              void* d_out, int out_size, void* d_ws, size_t ws_size,
                              hipStream_t stream) {
    const float*         pd      = (const float*)d_in[0];
    const float*         gt      = (const float*)d_in[1];
    const unsigned char* pd_mask = (const unsigned char*)d_in[2]; // JAX bool = 1 byte
    const unsigned char* gt_mask = (const unsigned char*)d_in[3];
    float* out      = (float*)d_out;
    float* partials = (float*)d_ws; // needs NBLOCKS * 4 = 2 KB

    mse_main_kernel<<<NBLOCKS, NTHREADS, 0, stream>>>(pd, gt, pd_mask, gt_mask, partials);
    mse_finish_kernel<<<1, 512, 0, stream>>>(partials, out);
}